// GATEncoder_79920751444077
// MI455X (gfx1250) — compile-verified
//
#include <hip/hip_runtime.h>

typedef __attribute__((ext_vector_type(2))) float v2f;
typedef __attribute__((ext_vector_type(8))) float v8f;

#define NEG_SLOPE 0.2f
#define EPSV 1e-16f
#define TPB 256

// ---- order-preserving float <-> uint encoding for atomicMax-based segment max ----
__device__ __forceinline__ unsigned ord_f32(float f) {
  unsigned u = __float_as_uint(f);
  return (u & 0x80000000u) ? ~u : (u | 0x80000000u);
}
__device__ __forceinline__ float unord_f32(unsigned u) {
  return __uint_as_float((u & 0x80000000u) ? (u & 0x7FFFFFFFu) : ~u);
}

// ---- generic fill (also used to set float 0.0 and the -inf-ordinal 0u) ----
__global__ void fill_u32(unsigned* __restrict__ p, unsigned v, int n) {
  int i = blockIdx.x * blockDim.x + threadIdx.x;
  if (i < n) p[i] = v;
}

// ---- WMMA GEMM: C[N,64] = A[N,K] @ B[K,64], fp32, one wave per 16-row tile ----
// A stride == K, B row-major [K,64], N must be a multiple of 16 (100000 = 6250*16).
template <int K>
__global__ __launch_bounds__(32) void wmma_gemm_n64(const float* __restrict__ A,
                                                    const float* __restrict__ B,
                                                    float* __restrict__ C) {
  const int row0 = blockIdx.x * 16;
  const int lane = threadIdx.x;       // 0..31 (wave32)
  const bool hi  = lane >= 16;
  const int l    = lane & 15;

  v8f acc[4] = {v8f{}, v8f{}, v8f{}, v8f{}};

#pragma unroll
  for (int k0 = 0; k0 < K; k0 += 4) {
    // A fragment (16x4 f32): lanes 0-15 -> K = k0,k0+1 ; lanes 16-31 -> K = k0+2,k0+3
    const int ka = k0 + (hi ? 2 : 0);
    v2f a;
    a.x = A[(size_t)(row0 + l) * K + ka];
    a.y = A[(size_t)(row0 + l) * K + ka + 1];
#pragma unroll
    for (int c = 0; c < 4; ++c) {
      // B fragment (4x16 f32), column tile c
      const int col = c * 16 + l;
      v2f b;
      b.x = B[(size_t)ka * 64 + col];
      b.y = B[(size_t)(ka + 1) * 64 + col];
      acc[c] = __builtin_amdgcn_wmma_f32_16x16x4_f32(false, a, false, b,
                                                     (short)0, acc[c], false, false);
    }
  }
  // C/D layout: VGPR j, lanes 0-15 -> M=j, lanes 16-31 -> M=j+8
#pragma unroll
  for (int c = 0; c < 4; ++c)
#pragma unroll
    for (int j = 0; j < 8; ++j) {
      const int r = row0 + j + (hi ? 8 : 0);
      C[(size_t)r * 64 + c * 16 + l] = acc[c][j];
    }
}

// ---- per-node attention dots: a_s[n,h] = sum_d h[n,h*D+d]*att_s[h*D+d] ----
template <int H>
__global__ void att_dots(const float* __restrict__ hm, const float* __restrict__ att_s,
                         const float* __restrict__ att_d, float* __restrict__ as,
                         float* __restrict__ ad, int n) {
  int i = blockIdx.x * blockDim.x + threadIdx.x;
  if (i >= n) return;
  const int D = 64 / H;
#pragma unroll
  for (int h = 0; h < H; ++h) {
    float ss = 0.f, dd = 0.f;
#pragma unroll
    for (int d = 0; d < D; ++d) {
      float v = hm[(size_t)i * 64 + h * D + d];
      ss += v * att_s[h * D + d];
      dd += v * att_d[h * D + d];
    }
    as[(size_t)i * H + h] = ss;
    ad[(size_t)i * H + h] = dd;
  }
}

__device__ __forceinline__ void edge_nodes(const int* __restrict__ ei, int t, int E,
                                           int& s, int& d) {
  if (t < E) { s = ei[t]; d = ei[E + t]; }
  else       { s = t - E; d = s; }          // self loops appended after real edges
}

// ---- pass 1: segment max of leaky-relu logits ----
template <int H>
__global__ void edge_max(const int* __restrict__ ei, int E, int ET,
                         const float* __restrict__ as, const float* __restrict__ ad,
                         unsigned* __restrict__ m) {
  int t = blockIdx.x * blockDim.x + threadIdx.x;
  if (t >= ET) return;
  int s, d; edge_nodes(ei, t, E, s, d);
#pragma unroll
  for (int h = 0; h < H; ++h) {
    float e = as[(size_t)s * H + h] + ad[(size_t)d * H + h];
    e = e > 0.f ? e : NEG_SLOPE * e;
    atomicMax(&m[(size_t)d * H + h], ord_f32(e));
  }
}

// ---- pass 2: segment sum of exp(logit - max) ----
template <int H>
__global__ void edge_sum(const int* __restrict__ ei, int E, int ET,
                         const float* __restrict__ as, const float* __restrict__ ad,
                         const unsigned* __restrict__ m, float* __restrict__ den) {
  int t = blockIdx.x * blockDim.x + threadIdx.x;
  if (t >= ET) return;
  int s, d; edge_nodes(ei, t, E, s, d);
#pragma unroll
  for (int h = 0; h < H; ++h) {
    float e = as[(size_t)s * H + h] + ad[(size_t)d * H + h];
    e = e > 0.f ? e : NEG_SLOPE * e;
    atomicAdd(&den[(size_t)d * H + h], __expf(e - unord_f32(m[(size_t)d * H + h])));
  }
}

// ---- pass 3: 64 lanes per edge; lane c handles feature column c (coalesced row move) ----
template <int H>
__global__ void edge_scatter(const int* __restrict__ ei, int E, int ET,
                             const float* __restrict__ as, const float* __restrict__ ad,
                             const unsigned* __restrict__ m, const float* __restrict__ den,
                             const float* __restrict__ hsrc, float* __restrict__ acc) {
  int g = blockIdx.x * blockDim.x + threadIdx.x;
  int t = g >> 6;
  if (t >= ET) return;
  int c = g & 63;
  int h = c / (64 / H);
  int s, d; edge_nodes(ei, t, E, s, d);
  float e = as[(size_t)s * H + h] + ad[(size_t)d * H + h];
  e = e > 0.f ? e : NEG_SLOPE * e;
  float alpha = __expf(e - unord_f32(m[(size_t)d * H + h])) / (den[(size_t)d * H + h] + EPSV);
  atomicAdd(&acc[(size_t)d * 64 + c], alpha * hsrc[(size_t)s * 64 + c]);
}

// ---- bias + ELU (layer1 activation), in place ----
__global__ void bias_elu(float* __restrict__ acc, const float* __restrict__ b, int total) {
  int i = blockIdx.x * blockDim.x + threadIdx.x;
  if (i >= total) return;
  float v = acc[i] + b[i & 63];
  acc[i] = v > 0.f ? v : (__expf(v) - 1.f);
}

// ---- final bias add (layer2; H2=1 so head-mean is identity) ----
__global__ void add_bias(float* __restrict__ out, const float* __restrict__ b, int total) {
  int i = blockIdx.x * blockDim.x + threadIdx.x;
  if (i >= total) return;
  out[i] += b[i & 63];
}

static inline int ceil_div(long long a, int b) { return (int)((a + b - 1) / b); }

extern "C" void kernel_launch(void* const* d_in, const int* in_sizes, int n_in,
                              void* d_out, int out_size, void* d_ws, size_t ws_size,
                              hipStream_t stream) {
  const float* x        = (const float*)d_in[0];
  const int*   ei       = (const int*)d_in[1];
  const float* W1       = (const float*)d_in[2];
  const float* att_src1 = (const float*)d_in[3];
  const float* att_dst1 = (const float*)d_in[4];
  const float* b1       = (const float*)d_in[5];
  const float* W2       = (const float*)d_in[6];
  const float* att_src2 = (const float*)d_in[7];
  const float* att_dst2 = (const float*)d_in[8];
  const float* b2       = (const float*)d_in[9];
  float* out = (float*)d_out;

  const int N  = in_sizes[0] / 8;   // 100000
  const int E  = in_sizes[1] / 2;   // 1600000
  const int ET = E + N;             // with self loops

  // workspace layout (layer2 reuses layer1's dead buffers)
  float*    h1   = (float*)d_ws;                          // N*64
  float*    acc1 = h1 + (size_t)N * 64;                   // N*64 (becomes ELU output)
  float*    as1  = acc1 + (size_t)N * 64;                 // N*8
  float*    ad1  = as1 + (size_t)N * 8;                   // N*8
  unsigned* m1   = (unsigned*)(ad1 + (size_t)N * 8);      // N*8
  float*    den1 = (float*)(m1 + (size_t)N * 8);          // N*8
  float*    h2   = h1;                                    // reuse
  float*    as2  = as1;
  float*    ad2  = ad1;
  unsigned* m2   = m1;
  float*    den2 = den1;

  // ---- init (ordinal 0u == most-negative for the max encoding; 0u == 0.0f) ----
  fill_u32<<<ceil_div((long long)N * 64, TPB), TPB, 0, stream>>>((unsigned*)acc1, 0u, N * 64);
  fill_u32<<<ceil_div((long long)N * 8,  TPB), TPB, 0, stream>>>(m1, 0u, N * 8);
  fill_u32<<<ceil_div((long long)N * 8,  TPB), TPB, 0, stream>>>((unsigned*)den1, 0u, N * 8);
  fill_u32<<<ceil_div((long long)N * 64, TPB), TPB, 0, stream>>>((unsigned*)out, 0u, N * 64);

  // ---- layer 1 ----
  wmma_gemm_n64<8><<<N / 16, 32, 0, stream>>>(x, W1, h1);
  att_dots<8><<<ceil_div(N, TPB), TPB, 0, stream>>>(h1, att_src1, att_dst1, as1, ad1, N);
  edge_max<8><<<ceil_div(ET, TPB), TPB, 0, stream>>>(ei, E, ET, as1, ad1, m1);
  edge_sum<8><<<ceil_div(ET, TPB), TPB, 0, stream>>>(ei, E, ET, as1, ad1, m1, den1);
  edge_scatter<8><<<ceil_div((long long)ET * 64, TPB), TPB, 0, stream>>>(
      ei, E, ET, as1, ad1, m1, den1, h1, acc1);
  bias_elu<<<ceil_div((long long)N * 64, TPB), TPB, 0, stream>>>(acc1, b1, N * 64);

  // ---- layer 2 (re-init reused softmax buffers) ----
  fill_u32<<<ceil_div(N, TPB), TPB, 0, stream>>>(m2, 0u, N);
  fill_u32<<<ceil_div(N, TPB), TPB, 0, stream>>>((unsigned*)den2, 0u, N);

  wmma_gemm_n64<64><<<N / 16, 32, 0, stream>>>(acc1, W2, h2);
  att_dots<1><<<ceil_div(N, TPB), TPB, 0, stream>>>(h2, att_src2, att_dst2, as2, ad2, N);
  edge_max<1><<<ceil_div(ET, TPB), TPB, 0, stream>>>(ei, E, ET, as2, ad2, m2);
  edge_sum<1><<<ceil_div(ET, TPB), TPB, 0, stream>>>(ei, E, ET, as2, ad2, m2, den2);
  edge_scatter<1><<<ceil_div((long long)ET * 64, TPB), TPB, 0, stream>>>(
      ei, E, ET, as2, ad2, m2, den2, h2, out);
  add_bias<<<ceil_div((long long)N * 64, TPB), TPB, 0, stream>>>(out, b2, N * 64);
}